// DynamicUpsamplingFilter_3C_1297080123483
// MI455X (gfx1250) — compile-verified
//
#include <hip/hip_runtime.h>

typedef float v2f __attribute__((ext_vector_type(2)));
typedef float v8f __attribute__((ext_vector_type(8)));

constexpr int Bn   = 8;
constexpr int Cn   = 3;
constexpr int Hn   = 180;
constexpr int Wn   = 320;
constexpr int NF   = 25;   // 5x5 patch taps
constexpr int Rn   = 16;
constexpr int PADn = 2;
constexpr int TILE_W = 32;
constexpr int LS   = 33;   // LDS row stride in floats; odd -> conflict-free gathers
constexpr int HW   = Hn * Wn;

__global__ __launch_bounds__(256) void duf_wmma_kernel(const float* __restrict__ x,
                                                       const float* __restrict__ filt,
                                                       float* __restrict__ out)
{
    // Filters for 32 pixels: 400 rows (indexed r*25+f) x 32, stride 33 -> 52.8 KB
    __shared__ float sF[NF * Rn * LS];
    // Output staging: 48 channel rows x 32 pixels -> 6.3 KB
    __shared__ float sO[Cn * Rn * LS];

    const int ntw  = Wn / TILE_W;                 // 10
    const int tile = blockIdx.x;
    const int tw   = tile % ntw;
    const int th   = (tile / ntw) % Hn;
    const int b    = tile / (ntw * Hn);
    const int w0   = tw * TILE_W;
    const int t    = threadIdx.x;

    // ---- Stage filters: coalesced 128B rows, lanes over w ----
    const float* fbase = filt + (size_t)b * NF * Rn * HW + (size_t)th * Wn + w0;
    for (int idx = t; idx < NF * Rn * TILE_W; idx += 256) {
        const int rowg = idx >> 5;                // f*16 + r (global layout)
        const int ww   = idx & 31;
        const int f    = rowg >> 4;
        const int r    = rowg & 15;
        sF[(r * NF + f) * LS + ww] = fbase[(size_t)rowg * HW + ww];
    }
    __syncthreads();

    const int  lane = t & 31;
    const int  wave = t >> 5;
    const int  cL   = lane & 15;                  // N-lane index (= r for B, = c row for A)
    const bool hi   = lane >= 16;
    const int  cx   = (cL < Cn) ? cL : (Cn - 1); // clamped channel for safe loads
    const bool cvalid = (cL < Cn);

    for (int i = 0; i < 4; ++i) {
        const int p  = wave * 4 + i;              // pixel within the 32-wide tile
        const int wp = w0 + p;

        // Per-lane patch values for this lane's channel (zero outside valid rows).
        float patch[28];
        const float* xc = x + (size_t)(b * Cn + cx) * HW;
        #pragma unroll
        for (int f = 0; f < NF; ++f) {
            const int di = f / 5, dj = f % 5;
            const int hh = th + di - PADn;
            const int ww = wp + dj - PADn;
            const int hcl = hh < 0 ? 0 : (hh >= Hn ? Hn - 1 : hh);
            const int wcl = ww < 0 ? 0 : (ww >= Wn ? Wn - 1 : ww);
            const float v = xc[(size_t)hcl * Wn + wcl];
            const bool ok = cvalid && (hh >= 0) && (hh < Hn) && (ww >= 0) && (ww < Wn);
            patch[f] = ok ? v : 0.0f;
        }
        patch[25] = 0.0f; patch[26] = 0.0f; patch[27] = 0.0f;

        // D[c][r] += sum_f patch[c][f] * filt[f][r], K padded 25 -> 28 (7 chunks of 4)
        v8f acc = {};
        #pragma unroll
        for (int kk = 0; kk < 7; ++kk) {
            const int f0 = 4 * kk;
            v2f a, bm;
            // A 16x4 f32: VGPR0 = K0 (lanes 0-15) / K2 (lanes 16-31), VGPR1 = K1 / K3
            a.x = hi ? patch[f0 + 2] : patch[f0 + 0];
            a.y = hi ? patch[f0 + 3] : patch[f0 + 1];
            // B 4x16 f32: same K striping; N = cL. Bank-conflict-free (odd LS).
            const int fb0 = f0 + (hi ? 2 : 0);
            const int fb1 = fb0 + 1;
            const int fc0 = fb0 < NF ? fb0 : NF - 1;
            const int fc1 = fb1 < NF ? fb1 : NF - 1;
            const float b0 = sF[(cL * NF + fc0) * LS + p];
            const float b1 = sF[(cL * NF + fc1) * LS + p];
            bm.x = (fb0 < NF) ? b0 : 0.0f;
            bm.y = (fb1 < NF) ? b1 : 0.0f;
            acc = __builtin_amdgcn_wmma_f32_16x16x4_f32(false, a, false, bm,
                                                        (short)0, acc, false, false);
        }

        // D rows M=0..2 live in acc[0..2], lanes 0-15 (N = r = lane).
        if (lane < 16) {
            sO[(0 * Rn + lane) * LS + p] = acc[0];
            sO[(1 * Rn + lane) * LS + p] = acc[1];
            sO[(2 * Rn + lane) * LS + p] = acc[2];
        }
    }
    __syncthreads();

    // ---- Coalesced output: 48 channel rows of 32 contiguous floats ----
    float* obase = out + (size_t)b * (Cn * Rn) * HW + (size_t)th * Wn + w0;
    for (int idx = t; idx < Cn * Rn * TILE_W; idx += 256) {
        const int ch = idx >> 5;
        const int ww = idx & 31;
        obase[(size_t)ch * HW + ww] = sO[ch * LS + ww];
    }
}

extern "C" void kernel_launch(void* const* d_in, const int* in_sizes, int n_in,
                              void* d_out, int out_size, void* d_ws, size_t ws_size,
                              hipStream_t stream) {
    const float* x       = (const float*)d_in[0];
    const float* filters = (const float*)d_in[1];
    float* out           = (float*)d_out;
    (void)in_sizes; (void)n_in; (void)out_size; (void)d_ws; (void)ws_size;

    const int nblocks = Bn * Hn * (Wn / TILE_W);  // 8*180*10 = 14400
    duf_wmma_kernel<<<dim3(nblocks), dim3(256), 0, stream>>>(x, filters, out);
}